// GroupDropout_49950469652647
// MI455X (gfx1250) — compile-verified
//
#include <hip/hip_runtime.h>
#include <stdint.h>

#define GROUPS 64
#define GROUP_SIZE 64
#define FEATS 4096
#define ROWS_PER_BLOCK 16
#define COLS_PER_BLOCK 1024  // 256 threads * 4 floats

typedef __attribute__((ext_vector_type(4))) unsigned int v4u;
typedef __attribute__((ext_vector_type(8))) int v8i;
typedef __attribute__((ext_vector_type(4))) int v4i;

#if __has_builtin(__builtin_amdgcn_tensor_load_to_lds)
#define USE_TDM 1
#else
#define USE_TDM 0
#endif

#if USE_TDM
// 1D TDM copy: n fp32 elements from global gsrc -> LDS byte offset lds_off.
// D# group0: count=1 | lds_addr | global_addr | type=2.
// D# group1: data_size=4B, tensor_dim0=tile_dim0=stride=n, tensor_dim1=1,
//            no multicast / iterate / pad / atomic-barrier.
__device__ __forceinline__ void tdm_load_1d_f32(const void* gsrc, unsigned lds_off,
                                                unsigned n) {
  unsigned long long ga = (unsigned long long)(uintptr_t)gsrc;
  v4u g0;
  g0[0] = 1u;                                                // count=1, user desc
  g0[1] = lds_off;                                           // lds_addr (bytes)
  g0[2] = (unsigned)(ga & 0xFFFFFFFFu);                      // global_addr[31:0]
  g0[3] = (unsigned)((ga >> 32) & 0x01FFFFFFu) | (2u << 30); // addr[56:32] | type=2
  v8i g1;
  g1[0] = (int)(2u << 16);                              // wg_mask=0, data_size=2 (4B)
  g1[1] = (int)((n & 0xFFFFu) << 16);                   // tensor_dim0[15:0]  (63:48)
  g1[2] = (int)(((n >> 16) & 0xFFFFu) | (1u << 16));    // dim0[31:16] | tensor_dim1=1
  g1[3] = (int)((n & 0xFFFFu) << 16);                   // tile_dim0 (127:112)
  g1[4] = 0;                                            // tile_dim1=0, tile_dim2=0
  g1[5] = (int)n;                                       // tensor_dim0_stride[31:0]
  g1[6] = 0;
  g1[7] = 0;
  v4i g2 = {0, 0, 0, 0};
  v4i g3 = {0, 0, 0, 0};
#if defined(__clang_major__) && (__clang_major__ >= 23)
  v8i g4 = {0, 0, 0, 0, 0, 0, 0, 0};
  __builtin_amdgcn_tensor_load_to_lds(g0, g1, g2, g3, g4, 0);
#else
  __builtin_amdgcn_tensor_load_to_lds(g0, g1, g2, g3, 0);
#endif
}
#endif

// Main kernel (defined first so it leads the device asm): out = x * feat_mask
// broadcast over rows. Each block owns a 16-row x 1024-col tile; the mask chunk
// is staged into LDS by the Tensor Data Mover, read once per thread as float4,
// and reused across 16 rows of b128 global traffic.
__global__ __launch_bounds__(256) void gd_apply_kernel(
    const float* __restrict__ x, const float* __restrict__ feat_mask,
    float* __restrict__ out, int nrows) {
  __shared__ float smask[COLS_PER_BLOCK];  // 4 KB
  const int tid = (int)threadIdx.x;
  const int chunk = (int)blockIdx.x & 3;  // FEATS / COLS_PER_BLOCK == 4
  const int rowTile = (int)blockIdx.x >> 2;
  const int col = chunk * COLS_PER_BLOCK + tid * 4;

  float4 m;
#if USE_TDM
  if (tid < 32) {  // wave 0 issues the DMA (TDM ignores EXEC; per-wave issue)
    tdm_load_1d_f32(feat_mask + chunk * COLS_PER_BLOCK,
                    (unsigned)(uintptr_t)(void*)smask, COLS_PER_BLOCK);
    __builtin_amdgcn_s_wait_tensorcnt(0);
  }
  __syncthreads();
  asm volatile("" ::: "memory");  // LDS written by TDM behind the compiler's back
  m = *reinterpret_cast<const float4*>(&smask[tid * 4]);
#else
  m = *reinterpret_cast<const float4*>(feat_mask + col);
  (void)smask;
#endif

  const int row0 = rowTile * ROWS_PER_BLOCK;
  int rmax = nrows - row0;
  if (rmax > ROWS_PER_BLOCK) rmax = ROWS_PER_BLOCK;

  const float* xp = x + (size_t)row0 * FEATS + col;
  float* op = out + (size_t)row0 * FEATS + col;
#pragma unroll 4
  for (int r = 0; r < rmax; ++r) {
    __builtin_prefetch(xp + (size_t)4 * FEATS, 0, 1);
    float4 v = *reinterpret_cast<const float4*>(xp);
    float4 o = make_float4(v.x * m.x, v.y * m.y, v.z * m.z, v.w * m.w);
    *reinterpret_cast<float4*>(op) = o;
    xp += FEATS;
    op += FEATS;
  }
}

__device__ __forceinline__ unsigned rotl32(unsigned v, int r) {
  return (v << r) | (v >> (32 - r));
}

// JAX classic threefry2x32: key=(k0,k1), counter pair (c0,c1), 20 rounds.
__device__ __forceinline__ void threefry2x32(unsigned k0, unsigned k1,
                                             unsigned c0, unsigned c1,
                                             unsigned* o0, unsigned* o1) {
  const unsigned ks2 = k0 ^ k1 ^ 0x1BD11BDAu;
  unsigned v0 = c0 + k0;
  unsigned v1 = c1 + k1;
  const int rotA[4] = {13, 15, 26, 6};
  const int rotB[4] = {17, 29, 16, 24};
#pragma unroll
  for (int i = 0; i < 4; ++i) { v0 += v1; v1 = rotl32(v1, rotA[i]); v1 ^= v0; }
  v0 += k1;  v1 += ks2 + 1u;
#pragma unroll
  for (int i = 0; i < 4; ++i) { v0 += v1; v1 = rotl32(v1, rotB[i]); v1 ^= v0; }
  v0 += ks2; v1 += k0 + 2u;
#pragma unroll
  for (int i = 0; i < 4; ++i) { v0 += v1; v1 = rotl32(v1, rotA[i]); v1 ^= v0; }
  v0 += k0;  v1 += k1 + 3u;
#pragma unroll
  for (int i = 0; i < 4; ++i) { v0 += v1; v1 = rotl32(v1, rotB[i]); v1 ^= v0; }
  v0 += k1;  v1 += ks2 + 4u;
#pragma unroll
  for (int i = 0; i < 4; ++i) { v0 += v1; v1 = rotl32(v1, rotA[i]); v1 ^= v0; }
  v0 += ks2; v1 += k0 + 5u;
  *o0 = v0; *o1 = v1;
}

// Setup kernel: 64 group keep/drop decisions (threefry2x32, key 42, p=0.2),
// scattered through `groups` into feat_mask[4096] held in d_ws.
__global__ __launch_bounds__(64) void gd_mask_kernel(
    const int* __restrict__ groups, float* __restrict__ feat_mask) {
  __shared__ float lmask[FEATS];
  const int g = (int)threadIdx.x;  // 0..63
  for (int j = g; j < FEATS; j += GROUPS) lmask[j] = 1.0f;

  // counts = iota(64); legacy split: pair (i, i+32); word0 for i<32 else word1
  const unsigned c = (unsigned)(g & 31);
  unsigned o0, o1;
  threefry2x32(0u, 42u, c, c + 32u, &o0, &o1);
  const unsigned bits = (g < 32) ? o0 : o1;
  const float u = __uint_as_float((bits >> 9) | 0x3F800000u) - 1.0f;  // [0,1)
  const float keep = (u < 0.2f) ? 0.0f : 1.0f;

  __syncthreads();
  const int base = g * GROUP_SIZE;
#pragma unroll 8
  for (int j = 0; j < GROUP_SIZE; ++j) lmask[groups[base + j]] = keep;
  __syncthreads();
  for (int j = g; j < FEATS; j += GROUPS) feat_mask[j] = lmask[j];
}

extern "C" void kernel_launch(void* const* d_in, const int* in_sizes, int n_in,
                              void* d_out, int out_size, void* d_ws, size_t ws_size,
                              hipStream_t stream) {
  (void)n_in; (void)out_size; (void)ws_size;
  const float* x = (const float*)d_in[0];
  const int* groups = (const int*)d_in[1];  // jnp int64 -> int32 under default x64-off
  float* out = (float*)d_out;
  float* feat_mask = (float*)d_ws;  // 4096 floats of scratch

  const int B = in_sizes[0] / FEATS;  // 16384

  gd_mask_kernel<<<1, GROUPS, 0, stream>>>(groups, feat_mask);

  const int rowTiles = (B + ROWS_PER_BLOCK - 1) / ROWS_PER_BLOCK;
  gd_apply_kernel<<<dim3(4u * (unsigned)rowTiles), 256, 0, stream>>>(
      x, feat_mask, out, B);
}